// kw_rank_loss_32263794327816
// MI455X (gfx1250) — compile-verified
//
#include <hip/hip_runtime.h>
#include <math.h>

typedef __attribute__((ext_vector_type(2))) float v2f;
typedef __attribute__((ext_vector_type(8))) float v8f;

#define DDIM 256
#define NROWS 262144
#define NBLOCKS 2048
#define ROWS_PER_BLOCK (NROWS / NBLOCKS)   // 128
#define MARGIN_F 0.75f
#define ACC_LEN (4 * DDIM + 4)             // 4x256 sums + 4 counts

__global__ void kw_zero_acc(float* __restrict__ acc, int n) {
  int i = blockIdx.x * blockDim.x + threadIdx.x;
  if (i < n) acc[i] = 0.0f;
}

// Masked group-sum via V_WMMA_F32_16X16X4_F32.
//   A (16x4): A[g,k] = mask(group g, row n+k); only rows g=0..3 nonzero.
//   B (4x16): B[k,c] = X[n+k, col0+c]
//   C (16x16) accumulates sums[g, col] in fp32.
// A/B per-lane layout (32-bit 16x4 operands):
//   lanes 0-15 : v0 -> K=0, v1 -> K=1   (M/N = lane)
//   lanes 16-31: v0 -> K=2, v1 -> K=3   (M/N = lane-16)
__global__ __launch_bounds__(512) void kw_group_sum_wmma(
    const float* __restrict__ X, const float* __restrict__ T,
    float* __restrict__ acc)
{
  const int wave = threadIdx.x >> 5;       // 0..15 -> 16-column tile
  const int lane = threadIdx.x & 31;
  const int li   = lane & 15;
  const int half = lane >> 4;
  const int col  = (wave << 4) + li;
  const long rowStart = (long)blockIdx.x * ROWS_PER_BLOCK;

  // Per-lane group interval (lo, hi]; lanes with li>=4 get an empty interval.
  // Pure value selects -> v_cndmask chain, hoisted out of the hot loop.
  const float BIG = 3.0e38f;
  const float lo = (li == 0) ? -BIG
                 : (li == 1) ? 0.3f
                 : (li == 2) ? 0.6f
                 : (li == 3) ? 0.8f
                 :             BIG;        // empty
  const float hi = (li == 0) ? 0.1f
                 : (li == 1) ? 0.4f
                 : (li == 2) ? 0.7f
                 : (li == 3) ? 1.1f
                 :             -BIG;       // empty

  v8f c = {};
#pragma unroll 2
  for (int r = 0; r < ROWS_PER_BLOCK; r += 4) {
    const long n = rowStart + r + 2 * half;  // this lane covers rows n, n+1

    // targets for the two K-slots of this lane: contiguous pair -> b64 load
    const v2f tv = *(const v2f*)(T + n);

    v2f a, b;
    b.x = X[n * DDIM + col];
    b.y = X[(n + 1) * DDIM + col];
    // branchless membership: non-short-circuit & keeps it as v_cmp/s_and/v_cndmask
    a.x = ((tv.x > lo) & (tv.x <= hi)) ? 1.0f : 0.0f;
    a.y = ((tv.y > lo) & (tv.y <= hi)) ? 1.0f : 0.0f;

    // Unguarded speculative prefetch (ISA §10.5: translation failures are
    // silently dropped at SE scope) — keeps the hot loop branch-free.
    __builtin_prefetch(&X[(n + 16) * DDIM + col], 0, 3);

    // D = A x B + C  (fp32 WMMA, round-to-nearest-even)
    c = __builtin_amdgcn_wmma_f32_16x16x4_f32(
        /*neg_a=*/false, a, /*neg_b=*/false, b,
        /*c_mod=*/(short)0, c, /*reuse_a=*/false, /*reuse_b=*/false);
  }

  // C layout: VGPR j, lanes 0-15 hold M=j. Groups 0..3 live in c[0..3], lanes 0-15.
  if (half == 0) {
#pragma unroll
    for (int g = 0; g < 4; ++g)
      unsafeAtomicAdd(&acc[g * DDIM + col], c[g]);
  }

  // group counts: wave 0 only (tiny)
  if (wave == 0) {
    float cnt0 = 0.f, cnt1 = 0.f, cnt2 = 0.f, cnt3 = 0.f;
    for (int r = lane; r < ROWS_PER_BLOCK; r += 32) {
      float t = T[rowStart + r];
      cnt0 += (t <= 0.1f)                    ? 1.f : 0.f;
      cnt1 += ((t > 0.3f) & (t <= 0.4f))     ? 1.f : 0.f;
      cnt2 += ((t > 0.6f) & (t <= 0.7f))     ? 1.f : 0.f;
      cnt3 += ((t > 0.8f) & (t <= 1.1f))     ? 1.f : 0.f;
    }
#pragma unroll
    for (int off = 16; off > 0; off >>= 1) {
      cnt0 += __shfl_down(cnt0, off, 32);
      cnt1 += __shfl_down(cnt1, off, 32);
      cnt2 += __shfl_down(cnt2, off, 32);
      cnt3 += __shfl_down(cnt3, off, 32);
    }
    if (lane == 0) {
      unsafeAtomicAdd(&acc[4 * DDIM + 0], cnt0);
      unsafeAtomicAdd(&acc[4 * DDIM + 1], cnt1);
      unsafeAtomicAdd(&acc[4 * DDIM + 2], cnt2);
      unsafeAtomicAdd(&acc[4 * DDIM + 3], cnt3);
    }
  }
}

__device__ __forceinline__ float kw_relu(float x) { return x > 0.f ? x : 0.f; }

__global__ void kw_finalize_loss(const float* __restrict__ acc,
                                 float* __restrict__ out)
{
  if (threadIdx.x != 0 || blockIdx.x != 0) return;

  float cnt[4];
#pragma unroll
  for (int g = 0; g < 4; ++g) cnt[g] = acc[4 * DDIM + g];

  float s12 = 0.f, s13 = 0.f, s14 = 0.f, s23 = 0.f, s24 = 0.f, s34 = 0.f;
  for (int d = 0; d < DDIM; ++d) {
    float m[4];
#pragma unroll
    for (int g = 0; g < 4; ++g) {
      float cg = cnt[g];
      m[g] = (cg > 0.f) ? acc[g * DDIM + d] / fmaxf(cg, 1.f) : 0.f;
    }
    float e;
    e = m[0] - m[1]; s12 += e * e;
    e = m[0] - m[2]; s13 += e * e;
    e = m[0] - m[3]; s14 += e * e;
    e = m[1] - m[2]; s23 += e * e;
    e = m[1] - m[3]; s24 += e * e;
    e = m[2] - m[3]; s34 += e * e;
  }
  float d12 = sqrtf(s12), d13 = sqrtf(s13), d14 = sqrtf(s14);
  float d23 = sqrtf(s23), d24 = sqrtf(s24), d34 = sqrtf(s34);
  const float mg = MARGIN_F;
  float loss = kw_relu(d12 - d13 + mg)        + kw_relu(d12 - d14 + 2.f * mg)
             + kw_relu(d23 - d24 + mg)        + kw_relu(d23 - d14 + 2.f * mg)
             + kw_relu(d34 - d24 + mg)        + kw_relu(d34 - d14 + 2.f * mg);
  out[0] = loss;
}

extern "C" void kernel_launch(void* const* d_in, const int* in_sizes, int n_in,
                              void* d_out, int out_size, void* d_ws, size_t ws_size,
                              hipStream_t stream) {
  (void)in_sizes; (void)n_in; (void)out_size; (void)ws_size;

  const float* X = (const float*)d_in[0];   // [N, 256] fp32
  const float* T = (const float*)d_in[1];   // [N] fp32
  float* acc = (float*)d_ws;                // 4*256 sums + 4 counts
  float* out = (float*)d_out;

  kw_zero_acc<<<(ACC_LEN + 255) / 256, 256, 0, stream>>>(acc, ACC_LEN);
  kw_group_sum_wmma<<<NBLOCKS, 512, 0, stream>>>(X, T, acc);
  kw_finalize_loss<<<1, 32, 0, stream>>>(acc, out);
}